// JKNetReranker_48885317763306
// MI455X (gfx1250) — compile-verified
//
#include <hip/hip_runtime.h>
#include <math.h>

typedef __attribute__((ext_vector_type(2))) float v2f;
typedef __attribute__((ext_vector_type(8))) float v8f;

#define HID 128
#define LDS_STRIDE 132  // 128 + 4 pad: rows land on distinct banks

// ---------------- utility kernels ----------------

__global__ void zero_kernel(float* __restrict__ p, long n) {
    long i = (long)blockIdx.x * blockDim.x + threadIdx.x;
    long stride = (long)gridDim.x * blockDim.x;
    for (; i < n; i += stride) p[i] = 0.0f;
}

__global__ void count_edges_kernel(const int* __restrict__ ei, int n_edges,
                                   float* __restrict__ cnt) {
    int e = blockIdx.x * blockDim.x + threadIdx.x;
    if (e >= n_edges) return;
    int dst = ei[n_edges + e];
    atomicAdd(&cnt[dst], 1.0f);
}

__global__ void recip_kernel(float* __restrict__ c, int n) {
    int i = blockIdx.x * blockDim.x + threadIdx.x;
    if (i >= n) return;
    float v = c[i];
    c[i] = 1.0f / fmaxf(v, 1.0f);
}

// Pack weight matrix W[128][128] into row-pair float2 layout:
// P[(k/2)*128 + n] = { W[k][n], W[k+1][n] }  -> B fragment is one b64 load.
__global__ void pack_pairs_kernel(const float* __restrict__ W,
                                  float2* __restrict__ P) {
    int i = blockIdx.x * blockDim.x + threadIdx.x;  // 0 .. 64*128-1
    if (i >= (HID / 2) * HID) return;
    int kp = i >> 7;   // pair index
    int n  = i & 127;
    P[i] = make_float2(W[(2 * kp) * HID + n], W[(2 * kp + 1) * HID + n]);
}

// Per-edge gather/scatter: one wave (32 lanes) per edge, 4 floats per lane.
__global__ void scatter_kernel(const float* __restrict__ hin,
                               const int* __restrict__ ei, int n_edges,
                               float* __restrict__ agg) {
    long gid = (long)blockIdx.x * blockDim.x + threadIdx.x;
    int e = (int)(gid >> 5);
    int lane = (int)(gid & 31);
    if (e >= n_edges) return;
    int src = ei[e];
    int dst = ei[n_edges + e];
    const float4* hs = (const float4*)(hin + (long)src * HID);
    float4 v = hs[lane];
    float* ad = agg + (long)dst * HID + lane * 4;
    atomicAdd(ad + 0, v.x);
    atomicAdd(ad + 1, v.y);
    atomicAdd(ad + 2, v.z);
    atomicAdd(ad + 3, v.w);
}

// ---------------- fused SAGE GEMM (WMMA f32 16x16x4) ----------------
// hout[m][n] = relu( sum_k agg[m][k]*invc[m]*Wl[k][n] + sum_k hin[m][k]*Wr[k][n] + b[n] )
// One block = 16 output rows; 8 waves = 8 column tiles of 16.
// Two independent accumulator chains (one per GEMM) for matrix-pipe ILP.
__global__ void __launch_bounds__(256)
sage_gemm_kernel(const float* __restrict__ hin, const float* __restrict__ agg,
                 const float* __restrict__ invc,
                 const float2* __restrict__ Wlp, const float2* __restrict__ Wrp,
                 const float* __restrict__ bias,
                 float* __restrict__ hout, int n_nodes) {
    __shared__ float s_mean[16][LDS_STRIDE];
    __shared__ float s_h[16][LDS_STRIDE];

    int tid = threadIdx.x;
    int row0 = blockIdx.x * 16;

    // Stage: 256 threads, 16 rows x 128 cols, 8 floats (2 x float4) per thread.
    {
        int r = tid >> 4;          // 0..15
        int c = (tid & 15) * 8;    // 0..120
        long row = row0 + r;
        float inv = 1.0f;
        float4 a0 = make_float4(0.f, 0.f, 0.f, 0.f), a1 = a0, h0 = a0, h1 = a0;
        if (row < n_nodes) {
            inv = invc[row];
            const float4* ap = (const float4*)(agg + row * HID + c);
            const float4* hp = (const float4*)(hin + row * HID + c);
            a0 = ap[0]; a1 = ap[1];
            h0 = hp[0]; h1 = hp[1];
        }
        s_mean[r][c + 0] = a0.x * inv; s_mean[r][c + 1] = a0.y * inv;
        s_mean[r][c + 2] = a0.z * inv; s_mean[r][c + 3] = a0.w * inv;
        s_mean[r][c + 4] = a1.x * inv; s_mean[r][c + 5] = a1.y * inv;
        s_mean[r][c + 6] = a1.z * inv; s_mean[r][c + 7] = a1.w * inv;
        s_h[r][c + 0] = h0.x; s_h[r][c + 1] = h0.y;
        s_h[r][c + 2] = h0.z; s_h[r][c + 3] = h0.w;
        s_h[r][c + 4] = h1.x; s_h[r][c + 5] = h1.y;
        s_h[r][c + 6] = h1.z; s_h[r][c + 7] = h1.w;
    }
    __syncthreads();

    int wave = tid >> 5;            // 0..7 -> column tile
    int lane = tid & 31;
    int n0 = wave * 16;
    int m = lane & 15;              // row within tile (A operand)
    int koff = (lane >> 4) << 1;    // 0 for lanes 0-15, 2 for lanes 16-31
    int kpoff = lane >> 4;          // pair-index offset
    int n = n0 + (lane & 15);       // column (B/C/D operand)

    // Accumulator 0 seeded with bias; accumulator 1 starts at zero.
    float bv = bias[n];
    v8f acc0, acc1;
    #pragma unroll
    for (int i = 0; i < 8; ++i) { acc0[i] = bv; acc1[i] = 0.0f; }

    #pragma unroll 4
    for (int k = 0; k < HID; k += 4) {
        int kk = k + koff;
        int kp = (k >> 1) + kpoff;
        // GEMM 0: mean @ Wl  (independent chain)
        v2f a;
        a.x = s_mean[m][kk];
        a.y = s_mean[m][kk + 1];
        float2 bl = Wlp[kp * HID + n];
        v2f b; b.x = bl.x; b.y = bl.y;
        acc0 = __builtin_amdgcn_wmma_f32_16x16x4_f32(
            false, a, false, b, (short)0, acc0, false, false);

        // GEMM 1: h @ Wr  (independent chain)
        v2f a2;
        a2.x = s_h[m][kk];
        a2.y = s_h[m][kk + 1];
        float2 br = Wrp[kp * HID + n];
        v2f b2; b2.x = br.x; b2.y = br.y;
        acc1 = __builtin_amdgcn_wmma_f32_16x16x4_f32(
            false, a2, false, b2, (short)0, acc1, false, false);
    }

    // Combine, ReLU, store. C/D layout: VGPR i -> M=i (lanes 0-15) / M=i+8 (16-31).
    int mrow_off = (lane >> 4) << 3;   // +8 for upper half-wave
    if (row0 + 16 <= n_nodes) {
        #pragma unroll
        for (int i = 0; i < 8; ++i) {
            long row = row0 + i + mrow_off;
            hout[row * HID + n] = fmaxf(acc0[i] + acc1[i], 0.0f);
        }
    } else {
        #pragma unroll
        for (int i = 0; i < 8; ++i) {
            long row = row0 + i + mrow_off;
            if (row < n_nodes) hout[row * HID + n] = fmaxf(acc0[i] + acc1[i], 0.0f);
        }
    }
}

// score[node] += dot(h[node], Ws_l)   (one wave per node)
__global__ void score_kernel(const float* __restrict__ h,
                             const float* __restrict__ Wsl,
                             float* __restrict__ score, int n_nodes) {
    long gid = (long)blockIdx.x * blockDim.x + threadIdx.x;
    int node = (int)(gid >> 5);
    int lane = (int)(gid & 31);
    if (node >= n_nodes) return;
    const float* hr = h + (long)node * HID;
    float p = 0.0f;
    #pragma unroll
    for (int j = 0; j < HID; j += 32) p += hr[j + lane] * Wsl[j + lane];
    #pragma unroll
    for (int o = 16; o > 0; o >>= 1) p += __shfl_xor(p, o, 32);
    if (lane == 0) score[node] += p;
}

__global__ void finalize_kernel(const float* __restrict__ score,
                                const float* __restrict__ rer,
                                const float* __restrict__ bs,
                                const float* __restrict__ alpha,
                                float* __restrict__ out, int n_nodes) {
    int i = blockIdx.x * blockDim.x + threadIdx.x;
    if (i >= n_nodes) return;
    float a = 1.0f / (1.0f + expf(-alpha[0]));
    float gnn = score[i] + bs[0];
    out[i] = a * rer[i] + (1.0f - a) * gnn;
}

// ---------------- host orchestration ----------------

extern "C" void kernel_launch(void* const* d_in, const int* in_sizes, int n_in,
                              void* d_out, int out_size, void* d_ws, size_t ws_size,
                              hipStream_t stream) {
    const float* x     = (const float*)d_in[0];
    const int*   ei    = (const int*)d_in[1];
    const float* rer   = (const float*)d_in[2];
    const float* Wl[3] = {(const float*)d_in[3], (const float*)d_in[6], (const float*)d_in[9]};
    const float* Wr[3] = {(const float*)d_in[4], (const float*)d_in[7], (const float*)d_in[10]};
    const float* bb[3] = {(const float*)d_in[5], (const float*)d_in[8], (const float*)d_in[11]};
    const float* Ws    = (const float*)d_in[12];
    const float* bs    = (const float*)d_in[13];
    const float* alpha = (const float*)d_in[14];

    int n_nodes = in_sizes[2];          // reranker_scores: [N]
    int n_edges = in_sizes[1] / 2;      // edge_index: [2, E]
    long feat = (long)n_nodes * HID;
    const int WPACK = (HID / 2) * HID;  // float2 elements per packed matrix

    float* ws    = (float*)d_ws;
    float* agg   = ws;                  // [N,128]
    float* hA    = agg + feat;          // [N,128]
    float* hB    = hA + feat;           // [N,128]
    float* invc  = hB + feat;           // [N]
    float* score = invc + n_nodes;      // [N]
    float2* wpack = (float2*)(score + n_nodes);  // 6 packed weight matrices

    const int TB = 256;
    int zgrid = 2048;
    int pgrid = (WPACK + TB - 1) / TB;

    // zero invc + score (contiguous)
    zero_kernel<<<zgrid, TB, 0, stream>>>(invc, 2L * n_nodes);
    // in-degree counts -> reciprocals
    count_edges_kernel<<<(n_edges + TB - 1) / TB, TB, 0, stream>>>(ei, n_edges, invc);
    recip_kernel<<<(n_nodes + TB - 1) / TB, TB, 0, stream>>>(invc, n_nodes);

    // Pack the 6 weight matrices into row-pair layout.
    float2* Wlp[3]; float2* Wrp[3];
    for (int l = 0; l < 3; ++l) {
        Wlp[l] = wpack + (2 * l) * WPACK;
        Wrp[l] = wpack + (2 * l + 1) * WPACK;
        pack_pairs_kernel<<<pgrid, TB, 0, stream>>>(Wl[l], Wlp[l]);
        pack_pairs_kernel<<<pgrid, TB, 0, stream>>>(Wr[l], Wrp[l]);
    }

    const float* hin = x;
    float* houts[3] = {hA, hB, hA};
    long scat_threads = (long)n_edges * 32;
    int gemm_grid = (n_nodes + 15) / 16;

    for (int l = 0; l < 3; ++l) {
        zero_kernel<<<zgrid, TB, 0, stream>>>(agg, feat);
        scatter_kernel<<<(int)((scat_threads + TB - 1) / TB), TB, 0, stream>>>(
            hin, ei, n_edges, agg);
        sage_gemm_kernel<<<gemm_grid, TB, 0, stream>>>(
            hin, agg, invc, Wlp[l], Wrp[l], bb[l], houts[l], n_nodes);
        score_kernel<<<(int)(((long)n_nodes * 32 + TB - 1) / TB), TB, 0, stream>>>(
            houts[l], Ws + l * HID, score, n_nodes);
        hin = houts[l];
    }

    finalize_kernel<<<(n_nodes + TB - 1) / TB, TB, 0, stream>>>(
        score, rer, bs, alpha, (float*)d_out, n_nodes);
}